// TransformerBlock_pct_52725018526349
// MI455X (gfx1250) — compile-verified
//
#include <hip/hip_runtime.h>
#include <hip/hip_bf16.h>
#include <math.h>

// ---------------------------------------------------------------------------
// PCT offset-attention block on gfx1250 (MI455X), fp32 via V_WMMA_F32_16X16X4_F32
// B=16, N=4096, C=256
//
// GEMM mapping (wave32): wave-level 16x16 WMMA tiles, widened to
//   64x16 strips (row-space GEMMs) and 64x64 tiles (Gram matrix) so A/B
//   fragments are reused across subtiles -> fewer loads per v_wmma.
// ---------------------------------------------------------------------------

#define B_DIM 16
#define N_DIM 4096
#define C_DIM 256
#define ROWS  (B_DIM * N_DIM)     // 65536
#define BN_EPS 1e-5f
#define NCH 4                     // Gram-matrix reduction chunks
#define NCHUNK (N_DIM / NCH)      // 1024

typedef float v2f __attribute__((ext_vector_type(2)));
typedef float v8f __attribute__((ext_vector_type(8)));

// D = A(16x4) * B(4x16) + C, fp32 WMMA (codegen-confirmed on gfx1250).
__device__ __forceinline__ v8f wmma4(v2f a, v2f b, v8f c) {
    return __builtin_amdgcn_wmma_f32_16x16x4_f32(
        /*neg_a=*/false, a, /*neg_b=*/false, b,
        /*c_mod=*/(short)0, c, /*reuse_a=*/false, /*reuse_b=*/false);
}

// ---------------------------------------------------------------------------
// Kernel 1: K = X * Wq^T ; V = X * Wv^T + bv
// One wave computes a 64x16 strip of BOTH outputs: B-fragments (weights)
// reused across 4 row-subtiles, A-fragment reused across the two weights.
// Per K-step: 6 b64 loads for 8 wmma.
// ---------------------------------------------------------------------------
__global__ __launch_bounds__(128) void k_qv_gemm(
        const float* __restrict__ X,
        const float* __restrict__ Wq, const float* __restrict__ Wv,
        const float* __restrict__ bv,
        float* __restrict__ K, float* __restrict__ V) {
    const int lane = threadIdx.x & 31;
    const int wave = threadIdx.x >> 5;
    const int m  = lane & 15;
    const int kh = lane >> 4;           // which K-half of the fragment
    const int row0 = (blockIdx.x * 4 + wave) * 64;
    const int n0   = blockIdx.y * 16;

    const float* __restrict__ abase = X  + (size_t)(row0 + m) * C_DIM;
    const float* __restrict__ wqrow = Wq + (size_t)(n0  + m) * C_DIM;
    const float* __restrict__ wvrow = Wv + (size_t)(n0  + m) * C_DIM;

    const v8f vzero = {};
    v8f accK[4], accV[4];
    #pragma unroll
    for (int i = 0; i < 4; ++i) { accK[i] = vzero; accV[i] = vzero; }

    #pragma unroll 2
    for (int k = 0; k < C_DIM; k += 4) {
        const int off = k + 2 * kh;
        v2f bq, bw;
        bq[0] = wqrow[off];  bq[1] = wqrow[off + 1];
        bw[0] = wvrow[off];  bw[1] = wvrow[off + 1];
        #pragma unroll
        for (int i = 0; i < 4; ++i) {
            const float* __restrict__ ar = abase + (size_t)(16 * i) * C_DIM;
            v2f a;
            a[0] = ar[off];  a[1] = ar[off + 1];
            accK[i] = wmma4(a, bq, accK[i]);
            accV[i] = wmma4(a, bw, accV[i]);
        }
    }
    const float bias = bv[n0 + m];
    #pragma unroll
    for (int i = 0; i < 4; ++i) {
        #pragma unroll
        for (int e = 0; e < 8; ++e) {
            const size_t idx =
                (size_t)(row0 + 16 * i + e + 8 * kh) * C_DIM + n0 + m;
            K[idx] = accK[i][e];
            V[idx] = accV[i][e] + bias;
        }
    }
}

// ---------------------------------------------------------------------------
// Kernel 2: per-batch Gram matrix  E[b] = K[b]^T K[b]  (C x C, inner N)
// One wave computes a 64x64 tile (4x4 subtiles, 16 accumulators); the
// N-reduction is split into NCH chunks (partial buffers, summed in softmax
// kernel -> no float atomics, deterministic). A/B fragments shared across
// subtiles: 16 b32 loads per 16 wmma.
// ---------------------------------------------------------------------------
__global__ __launch_bounds__(32) void k_energy(
        const float* __restrict__ Kmat, float* __restrict__ Epart) {
    const int lane = threadIdx.x & 31;
    const int m  = lane & 15;
    const int kh = lane >> 4;
    const int c0 = blockIdx.x * 64;
    const int d0 = blockIdx.y * 64;
    const int b  = blockIdx.z / NCH;
    const int ch = blockIdx.z % NCH;

    const float* __restrict__ Kb =
        Kmat + (size_t)b * N_DIM * C_DIM + (size_t)ch * NCHUNK * C_DIM;

    const v8f vzero = {};
    v8f acc[4][4];
    #pragma unroll
    for (int i = 0; i < 4; ++i)
        #pragma unroll
        for (int j = 0; j < 4; ++j) acc[i][j] = vzero;

    for (int n = 0; n < NCHUNK; n += 4) {
        const float* __restrict__ p0 = Kb + (size_t)(n + 2 * kh) * C_DIM;
        const float* __restrict__ p1 = p0 + C_DIM;
        v2f av[4], bw[4];
        #pragma unroll
        for (int i = 0; i < 4; ++i) {
            av[i][0] = p0[c0 + 16 * i + m];
            av[i][1] = p1[c0 + 16 * i + m];
        }
        #pragma unroll
        for (int j = 0; j < 4; ++j) {
            bw[j][0] = p0[d0 + 16 * j + m];
            bw[j][1] = p1[d0 + 16 * j + m];
        }
        #pragma unroll
        for (int i = 0; i < 4; ++i)
            #pragma unroll
            for (int j = 0; j < 4; ++j)
                acc[i][j] = wmma4(av[i], bw[j], acc[i][j]);
    }

    float* __restrict__ Eo =
        Epart + ((size_t)ch * B_DIM + b) * C_DIM * C_DIM;
    #pragma unroll
    for (int i = 0; i < 4; ++i)
        #pragma unroll
        for (int j = 0; j < 4; ++j)
            #pragma unroll
            for (int e = 0; e < 8; ++e)
                Eo[(size_t)(c0 + 16 * i + e + 8 * kh) * C_DIM
                   + d0 + 16 * j + m] = acc[i][j][e];
}

// ---------------------------------------------------------------------------
// Kernel 3: attention = colrenorm( softmax_rows( sum_ch Epart ) )
// One block per batch; thread t owns row t (softmax) then column t (L1 sum);
// renorm factors broadcast via LDS so the final write is row-major.
// ---------------------------------------------------------------------------
__global__ __launch_bounds__(256) void k_softmax_renorm(
        const float* __restrict__ Epart, float* __restrict__ A) {
    const int b = blockIdx.x;
    const int t = threadIdx.x;
    const size_t bstride = (size_t)B_DIM * C_DIM * C_DIM;
    const float* __restrict__ Eb = Epart + (size_t)b * C_DIM * C_DIM;
    float* __restrict__ Ab = A + (size_t)b * C_DIM * C_DIM;

    // sum reduction chunks, track row max
    float mx = -INFINITY;
    for (int d = 0; d < C_DIM; ++d) {
        const size_t o = (size_t)t * C_DIM + d;
        float s = Eb[o];
        #pragma unroll
        for (int ch = 1; ch < NCH; ++ch) s += Eb[(size_t)ch * bstride + o];
        Ab[o] = s;
        mx = fmaxf(mx, s);
    }
    // row softmax
    float sum = 0.f;
    for (int d = 0; d < C_DIM; ++d) {
        const float e = expf(Ab[(size_t)t * C_DIM + d] - mx);
        Ab[(size_t)t * C_DIM + d] = e;
        sum += e;
    }
    const float inv = 1.f / sum;
    for (int d = 0; d < C_DIM; ++d) Ab[(size_t)t * C_DIM + d] *= inv;
    __threadfence();
    __syncthreads();

    // L1 column sums (axis=1), broadcast 1/colsum via LDS
    __shared__ float rcs[C_DIM];
    float cs = 0.f;
    for (int c = 0; c < C_DIM; ++c) cs += Ab[(size_t)c * C_DIM + t];
    rcs[t] = 1.f / (1e-9f + cs);
    __syncthreads();
    for (int d = 0; d < C_DIM; ++d)
        Ab[(size_t)t * C_DIM + d] *= rcs[d];
}

// ---------------------------------------------------------------------------
// Kernel 4: diff = X - V * A[b]     (64x16 strip per wave, B-frag shared)
// ---------------------------------------------------------------------------
__global__ __launch_bounds__(128) void k_xr_diff(
        const float* __restrict__ X, const float* __restrict__ V,
        const float* __restrict__ A, float* __restrict__ D) {
    const int lane = threadIdx.x & 31;
    const int wave = threadIdx.x >> 5;
    const int m  = lane & 15;
    const int kh = lane >> 4;
    const int row0 = (blockIdx.x * 4 + wave) * 64;
    const int d0   = blockIdx.y * 16;
    const int b    = row0 >> 12;                    // N_DIM == 4096

    const float* __restrict__ vbase = V + (size_t)(row0 + m) * C_DIM;
    const float* __restrict__ Ab    = A + (size_t)b * C_DIM * C_DIM;

    const v8f vzero = {};
    v8f acc[4];
    #pragma unroll
    for (int i = 0; i < 4; ++i) acc[i] = vzero;

    #pragma unroll 2
    for (int k = 0; k < C_DIM; k += 4) {
        const int off = k + 2 * kh;
        v2f bb;
        bb[0] = Ab[(size_t)off * C_DIM + d0 + m];
        bb[1] = Ab[(size_t)(off + 1) * C_DIM + d0 + m];
        #pragma unroll
        for (int i = 0; i < 4; ++i) {
            const float* __restrict__ vr = vbase + (size_t)(16 * i) * C_DIM;
            v2f a;
            a[0] = vr[off];  a[1] = vr[off + 1];
            acc[i] = wmma4(a, bb, acc[i]);
        }
    }
    #pragma unroll
    for (int i = 0; i < 4; ++i)
        #pragma unroll
        for (int e = 0; e < 8; ++e) {
            const size_t idx =
                (size_t)(row0 + 16 * i + e + 8 * kh) * C_DIM + d0 + m;
            D[idx] = X[idx] - acc[i][e];
        }
}

// ---------------------------------------------------------------------------
// Kernel 5: T = D * Wt^T + bt       (64x16 strip per wave)
// ---------------------------------------------------------------------------
__global__ __launch_bounds__(128) void k_t_gemm(
        const float* __restrict__ D, const float* __restrict__ Wt,
        const float* __restrict__ bt, float* __restrict__ T) {
    const int lane = threadIdx.x & 31;
    const int wave = threadIdx.x >> 5;
    const int m  = lane & 15;
    const int kh = lane >> 4;
    const int row0 = (blockIdx.x * 4 + wave) * 64;
    const int n0   = blockIdx.y * 16;

    const float* __restrict__ abase = D  + (size_t)(row0 + m) * C_DIM;
    const float* __restrict__ wtrow = Wt + (size_t)(n0  + m) * C_DIM;

    const v8f vzero = {};
    v8f acc[4];
    #pragma unroll
    for (int i = 0; i < 4; ++i) acc[i] = vzero;

    #pragma unroll 2
    for (int k = 0; k < C_DIM; k += 4) {
        const int off = k + 2 * kh;
        v2f bb;
        bb[0] = wtrow[off];  bb[1] = wtrow[off + 1];
        #pragma unroll
        for (int i = 0; i < 4; ++i) {
            const float* __restrict__ ar = abase + (size_t)(16 * i) * C_DIM;
            v2f a;
            a[0] = ar[off];  a[1] = ar[off + 1];
            acc[i] = wmma4(a, bb, acc[i]);
        }
    }
    const float bias = bt[n0 + m];
    #pragma unroll
    for (int i = 0; i < 4; ++i)
        #pragma unroll
        for (int e = 0; e < 8; ++e)
            T[(size_t)(row0 + 16 * i + e + 8 * kh) * C_DIM + n0 + m]
                = acc[i][e] + bias;
}

// ---------------------------------------------------------------------------
// Kernel 6a: coalesced per-band BN partial sums. Block g owns rows
// [256g, 256g+256); thread t = channel t -> consecutive addresses per load.
// ---------------------------------------------------------------------------
__global__ __launch_bounds__(256) void k_bn_partial(
        const float* __restrict__ T,
        float* __restrict__ Ps, float* __restrict__ Pq) {
    const int g = blockIdx.x;
    const int t = threadIdx.x;
    const int r0 = g * 256;
    float s = 0.f, q = 0.f;
    for (int r = 0; r < 256; ++r) {
        const float v = T[(size_t)(r0 + r) * C_DIM + t];
        s += v;
        q += v * v;
    }
    Ps[(size_t)g * C_DIM + t] = s;
    Pq[(size_t)g * C_DIM + t] = q;
}

// Kernel 6b: final per-channel mean / inv-std from 256 band partials.
__global__ __launch_bounds__(256) void k_bn_final(
        const float* __restrict__ Ps, const float* __restrict__ Pq,
        float* __restrict__ stats) {
    const int t = threadIdx.x;
    float s = 0.f, q = 0.f;
    for (int g = 0; g < ROWS / 256; ++g) {
        s += Ps[(size_t)g * C_DIM + t];
        q += Pq[(size_t)g * C_DIM + t];
    }
    const float inv_n = 1.f / (float)ROWS;
    const float mean  = s * inv_n;
    const float var   = q * inv_n - mean * mean;   // biased var
    stats[t]         = mean;
    stats[C_DIM + t] = rsqrtf(var + BN_EPS);
}

// ---------------------------------------------------------------------------
// Kernel 7: out = x + relu(gamma * (T - mean) * invstd + beta)
// ---------------------------------------------------------------------------
__global__ __launch_bounds__(256) void k_final(
        const float* __restrict__ X, const float* __restrict__ T,
        const float* __restrict__ stats,
        const float* __restrict__ gamma, const float* __restrict__ beta,
        float* __restrict__ out) {
    const size_t idx = (size_t)blockIdx.x * 256 + threadIdx.x;
    const int c = (int)(idx & (C_DIM - 1));
    const float th = (T[idx] - stats[c]) * stats[C_DIM + c];
    const float y  = gamma[c] * th + beta[c];
    out[idx] = X[idx] + fmaxf(y, 0.f);
}

// ---------------------------------------------------------------------------
// Launcher
// ---------------------------------------------------------------------------
extern "C" void kernel_launch(void* const* d_in, const int* in_sizes, int n_in,
                              void* d_out, int out_size, void* d_ws, size_t ws_size,
                              hipStream_t stream) {
    (void)in_sizes; (void)n_in; (void)out_size; (void)ws_size;
    // setup_inputs order: xyz, x, Wq, Wv, bv, Wt, bt, gamma, beta
    const float* X     = (const float*)d_in[1];
    const float* Wq    = (const float*)d_in[2];
    const float* Wv    = (const float*)d_in[3];
    const float* bv    = (const float*)d_in[4];
    const float* Wt    = (const float*)d_in[5];
    const float* bt    = (const float*)d_in[6];
    const float* gamma = (const float*)d_in[7];
    const float* beta  = (const float*)d_in[8];
    float* out = (float*)d_out;

    // Workspace layout (floats). diff reuses K, T reuses V (disjoint lifetimes).
    const size_t S  = (size_t)ROWS * C_DIM;          // 16M floats (64 MB)
    const size_t SE = (size_t)B_DIM * C_DIM * C_DIM; // 1M floats  (4 MB)
    float* ws    = (float*)d_ws;
    float* Kbuf  = ws;                 // [ROWS, C]          (later: diff)
    float* Vbuf  = ws + S;             // [ROWS, C]          (later: T)
    float* Epart = ws + 2 * S;         // [NCH, B, C, C] Gram partials
    float* Abuf  = Epart + NCH * SE;   // [B, C, C] attention
    float* Ps    = Abuf + SE;          // [256, C] BN partial sums
    float* Pq    = Ps + (ROWS / 256) * C_DIM;
    float* stats = Pq + (ROWS / 256) * C_DIM;        // [2, C]
    float* Dbuf  = Kbuf;
    float* Tbuf  = Vbuf;

    // 1) K = X Wq^T ; V = X Wv^T + bv
    k_qv_gemm<<<dim3(ROWS / 256, C_DIM / 16), 128, 0, stream>>>(X, Wq, Wv, bv, Kbuf, Vbuf);
    // 2) Gram partials: E[b] = K[b]^T K[b]
    k_energy<<<dim3(C_DIM / 64, C_DIM / 64, B_DIM * NCH), 32, 0, stream>>>(Kbuf, Epart);
    // 3) attention (chunk-sum + softmax + L1 column renorm)
    k_softmax_renorm<<<dim3(B_DIM), 256, 0, stream>>>(Epart, Abuf);
    // 4) diff = X - V A
    k_xr_diff<<<dim3(ROWS / 256, C_DIM / 16), 128, 0, stream>>>(X, Vbuf, Abuf, Dbuf);
    // 5) T = diff Wt^T + bt
    k_t_gemm<<<dim3(ROWS / 256, C_DIM / 16), 128, 0, stream>>>(Dbuf, Wt, bt, Tbuf);
    // 6) BN stats (coalesced two-stage)
    k_bn_partial<<<dim3(ROWS / 256), 256, 0, stream>>>(Tbuf, Ps, Pq);
    k_bn_final<<<dim3(1), 256, 0, stream>>>(Ps, Pq, stats);
    // 7) residual + BN + ReLU
    k_final<<<dim3((unsigned)(S / 256)), 256, 0, stream>>>(X, Tbuf, stats, gamma, beta, out);
}